// TimeAwareHSTU_5248450036146
// MI455X (gfx1250) — compile-verified
//
#include <hip/hip_runtime.h>
#include <hip/hip_bf16.h>

// ---------------- problem constants ----------------
#define Bsz    2
#define SEQ    2048
#define Hdim   1024
#define NH     8
#define HD     128
#define SIXH   6144     // 6*H
#define THREEH 3072     // 3*H
#define MTOT   4096     // Bsz*SEQ
#define INV_S  (1.0f / 2048.0f)

typedef float v2f __attribute__((ext_vector_type(2)));
typedef float v8f __attribute__((ext_vector_type(8)));
typedef int   v4i __attribute__((ext_vector_type(4)));

// D = A(16x4) * B(4x16) + C(16x16), fp32 WMMA (CDNA5 native fp32 matrix op)
__device__ __forceinline__ v8f wmma_f32(v2f a, v2f b, v8f c) {
    return __builtin_amdgcn_wmma_f32_16x16x4_f32(
        /*neg_a=*/false, a, /*neg_b=*/false, b,
        /*c_mod=*/(short)0, c, /*reuse_a=*/false, /*reuse_b=*/false);
}

__device__ __forceinline__ float silu_f(float x) { return x / (1.0f + __expf(-x)); }

// ---------------- async global->LDS staging (ASYNCcnt path) ----------------
#if __has_builtin(__builtin_amdgcn_global_load_async_to_lds_b128)
#define HAS_ASYNC_LDS 1
#else
#define HAS_ASYNC_LDS 0
#endif

__device__ __forceinline__ void tile_copy_b128(const float* g, float* l) {
#if HAS_ASYNC_LDS
    typedef __attribute__((address_space(1))) v4i gv4i;   // global (AS1)
    typedef __attribute__((address_space(3))) v4i lv4i;   // LDS    (AS3)
    __builtin_amdgcn_global_load_async_to_lds_b128(
        (gv4i*)(unsigned long long)(const void*)g,
        (lv4i*)(unsigned int)(unsigned long long)(void*)l,
        0, 0);
#else
    *(float4*)l = *(const float4*)g;
#endif
}

__device__ __forceinline__ void async_wait0() {
#if HAS_ASYNC_LDS
 #if __has_builtin(__builtin_amdgcn_s_wait_asynccnt)
    __builtin_amdgcn_s_wait_asynccnt(0);
 #else
    asm volatile("s_wait_asynccnt 0x0" ::: "memory");
 #endif
#endif
}

// =====================================================================
// Kernel 1: fused = silu(hidden @ W_qkvu)   (4096 x 1024) x (1024 x 6144)
// 4 waves/block stacked in M; each wave computes a 16x64 strip (4 WMMA
// tiles) reusing the A fragment across the 4 N-tiles. Block tile: 64x64.
// =====================================================================
__global__ void __launch_bounds__(128)
k_qkvu(const float* __restrict__ hs, const float* __restrict__ W,
       float* __restrict__ fused) {
    const int lane = threadIdx.x & 31;
    const int wave = threadIdx.x >> 5;
    const int half = lane >> 4;       // 0 or 1
    const int l16  = lane & 15;
    const int m0 = blockIdx.y * 64 + wave * 16;
    const int n0 = blockIdx.x * 64;

    const float* __restrict__ arow = hs + (size_t)(m0 + l16) * Hdim;

    v8f acc[4] = {};
    for (int kc = 0; kc < Hdim; kc += 64) {
        // prefetch next K-chunk (speculative; dropped if OOB)
        __builtin_prefetch(arow + kc + 64, 0, 1);
        __builtin_prefetch(W + (size_t)(kc + 64 + lane) * SIXH + n0, 0, 1);
        __builtin_prefetch(W + (size_t)(kc + 96 + lane) * SIXH + n0, 0, 1);
#pragma unroll 4
        for (int k0 = kc; k0 < kc + 64; k0 += 4) {
            const int ka = k0 + half * 2;
            v2f a;
            a.x = arow[ka];
            a.y = arow[ka + 1];
            const float* bp = W + (size_t)ka * SIXH + n0 + l16;
#pragma unroll
            for (int t = 0; t < 4; ++t) {
                v2f b;
                b.x = bp[t * 16];
                b.y = bp[(size_t)SIXH + t * 16];
                acc[t] = wmma_f32(a, b, acc[t]);
            }
        }
    }
#pragma unroll
    for (int t = 0; t < 4; ++t) {
        const int col = n0 + t * 16 + l16;
#pragma unroll
        for (int r = 0; r < 8; ++r) {
            const int row = m0 + half * 8 + r;
            fused[(size_t)row * SIXH + col] = silu_f(acc[t][r]);
        }
    }
}

// =====================================================================
// Kernel 2: attention core. One wave per (b, h, 16-row q tile).
// q/k tiles staged to LDS with async-to-LDS copies; RoPE computed from
// LDS; scores via WMMA; relu+mask+scale bounced through LDS to reshape
// D-layout -> A-layout; 3 accumulated AV GEMMs (rope, ts-bias, plain).
// fused columns: [0,3H)=gated  [3H,4H)=v  [4H,5H)=q  [5H,6H)=k
// =====================================================================
__global__ void __launch_bounds__(32)
k_attn(const float* __restrict__ fused, const int* __restrict__ am,
       const float* __restrict__ tsb, float* __restrict__ combg) {
    __shared__ float lq [16 * HD];   // raw q tile
    __shared__ float lqr[16 * HD];   // rope(q)
    __shared__ float lk [16 * HD];   // raw k tile
    __shared__ float lkr[16 * HD];   // rope(k)
    __shared__ float lsp[16 * 16];   // plain scores (A-layout staging)
    __shared__ float lsr[16 * 16];   // rope scores

    const int lane = threadIdx.x & 31;
    const int half = lane >> 4;
    const int l16  = lane & 15;
    const int q0 = blockIdx.x * 16;
    const int h  = blockIdx.y;
    const int bb = blockIdx.z;

    // ---- async-stage raw q tile (16 x 128 f32 = 8KB, 16B per lane-op) ----
#pragma unroll
    for (int it = 0; it < 16; ++it) {
        const int idx = (it * 32 + lane) * 4;     // float index in tile
        const int row = idx >> 7;
        const int col = idx & 127;
        const float* g = fused + ((size_t)(bb * SEQ + q0 + row)) * SIXH
                       + 4 * Hdim + h * HD + col;
        tile_copy_b128(g, &lq[idx]);
    }
    async_wait0();
    __syncthreads();

    // ---- rope(q) from LDS (interleaved-pair RoPE) ----
    for (int p = lane; p < 16 * 64; p += 32) {
        const int row = p >> 6;
        const int i   = p & 63;
        const float x0 = lq[row * HD + 2 * i];
        const float x1 = lq[row * HD + 2 * i + 1];
        const float invf = __powf(10000.0f, -(float)i / 64.0f);
        float sn, cs;
        __sincosf((float)(q0 + row) * invf, &sn, &cs);
        lqr[row * HD + 2 * i]     = x0 * cs - x1 * sn;
        lqr[row * HD + 2 * i + 1] = x1 * cs + x0 * sn;
    }
    __syncthreads();

    v8f acc_r[8], acc_t[8], acc_p[8];
#pragma unroll
    for (int nc = 0; nc < 8; ++nc) { acc_r[nc] = {}; acc_t[nc] = {}; acc_p[nc] = {}; }

    for (int mt = 0; mt < SEQ / 16; ++mt) {
        const int m0 = mt * 16;

        // ---- async-stage raw k tile ----
#pragma unroll
        for (int it = 0; it < 16; ++it) {
            const int idx = (it * 32 + lane) * 4;
            const int row = idx >> 7;
            const int col = idx & 127;
            const float* g = fused + ((size_t)(bb * SEQ + m0 + row)) * SIXH
                           + 5 * Hdim + h * HD + col;
            tile_copy_b128(g, &lk[idx]);
        }
        async_wait0();
        __syncthreads();

        // ---- rope(k) from LDS ----
        for (int p = lane; p < 16 * 64; p += 32) {
            const int row = p >> 6;
            const int i   = p & 63;
            const float x0 = lk[row * HD + 2 * i];
            const float x1 = lk[row * HD + 2 * i + 1];
            const float invf = __powf(10000.0f, -(float)i / 64.0f);
            float sn, cs;
            __sincosf((float)(m0 + row) * invf, &sn, &cs);
            lkr[row * HD + 2 * i]     = x0 * cs - x1 * sn;
            lkr[row * HD + 2 * i + 1] = x1 * cs + x0 * sn;
        }
        __syncthreads();

        // ---- scores: S = q . k^T over K = HD = 128 ----
        v8f sp = {}, sr = {};
        for (int k0 = 0; k0 < HD; k0 += 4) {
            const int ka = k0 + half * 2;
            v2f a, b;
            a.x = lq[l16 * HD + ka];  a.y = lq[l16 * HD + ka + 1];
            b.x = lk[l16 * HD + ka];  b.y = lk[l16 * HD + ka + 1];
            sp = wmma_f32(a, b, sp);
            a.x = lqr[l16 * HD + ka]; a.y = lqr[l16 * HD + ka + 1];
            b.x = lkr[l16 * HD + ka]; b.y = lkr[l16 * HD + ka + 1];
            sr = wmma_f32(a, b, sr);
        }

        // ---- relu * inv_s, mask, bounce D-layout -> A-layout via LDS ----
#pragma unroll
        for (int r = 0; r < 8; ++r) {
            const int n = r + half * 8;                    // local q row
            const size_t mi = ((size_t)bb * SEQ + q0 + n) * SEQ + m0 + l16;
            const int mv = am[mi];
            const float vp = sp[r] > 0.0f ? sp[r] * INV_S : 0.0f;
            const float vr = sr[r] > 0.0f ? sr[r] * INV_S : 0.0f;
            lsp[n * 16 + l16] = mv ? vp : 0.0f;
            lsr[n * 16 + l16] = mv ? vr : 0.0f;
        }
        __syncthreads();

        // ---- A-fragments for AV (hoisted; independent of column chunk) ----
        v2f ap4[4], ar4[4], at4[4];
#pragma unroll
        for (int kx = 0; kx < 4; ++kx) {
            const int ka = kx * 4 + half * 2;
            ap4[kx].x = lsp[l16 * 16 + ka]; ap4[kx].y = lsp[l16 * 16 + ka + 1];
            ar4[kx].x = lsr[l16 * 16 + ka]; ar4[kx].y = lsr[l16 * 16 + ka + 1];
            // masked ts-bias fragment: row n = q0+l16, col m = m0+ka
            const size_t tb = ((size_t)bb * SEQ + q0 + l16) * SEQ + m0 + ka;
            at4[kx].x = am[tb]     ? tsb[tb]     : 0.0f;
            at4[kx].y = am[tb + 1] ? tsb[tb + 1] : 0.0f;
        }

        // ---- AV: (16x16 scores) x (16x128 v), 8 column chunks ----
#pragma unroll
        for (int nc = 0; nc < 8; ++nc) {
#pragma unroll
            for (int kx = 0; kx < 4; ++kx) {
                const int ka = kx * 4 + half * 2;
                const float* vb = fused + ((size_t)(bb * SEQ + m0 + ka)) * SIXH
                                + 3 * Hdim + h * HD + nc * 16 + l16;
                v2f b; b.x = vb[0]; b.y = vb[SIXH];
                acc_p[nc] = wmma_f32(ap4[kx], b, acc_p[nc]);
                acc_r[nc] = wmma_f32(ar4[kx], b, acc_r[nc]);
                acc_t[nc] = wmma_f32(at4[kx], b, acc_t[nc]);
            }
        }
        __syncthreads();
    }

    // ---- write combined * gated : layout per token = [h][rope|ts|plain][d] ----
#pragma unroll
    for (int nc = 0; nc < 8; ++nc) {
#pragma unroll
        for (int r = 0; r < 8; ++r) {
            const int srow = q0 + r + half * 8;
            const int d = nc * 16 + l16;
            const size_t tok = (size_t)bb * SEQ + srow;
            const size_t c0 = (size_t)h * (3 * HD) + d;
            const float g0 = fused[tok * SIXH + c0];
            const float g1 = fused[tok * SIXH + c0 + HD];
            const float g2 = fused[tok * SIXH + c0 + 2 * HD];
            combg[tok * THREEH + c0]          = acc_r[nc][r] * g0;
            combg[tok * THREEH + c0 + HD]     = acc_t[nc][r] * g1;
            combg[tok * THREEH + c0 + 2 * HD] = acc_p[nc][r] * g2;
        }
    }
}

// =====================================================================
// Kernel 3: y = combg @ W_out + b_out + hidden   (4096 x 3072) x (3072 x 1024)
// Same 16x64-per-wave register blocking as kernel 1.
// =====================================================================
__global__ void __launch_bounds__(128)
k_outproj(const float* __restrict__ A, const float* __restrict__ W,
          const float* __restrict__ bias, const float* __restrict__ hs,
          float* __restrict__ y) {
    const int lane = threadIdx.x & 31;
    const int wave = threadIdx.x >> 5;
    const int half = lane >> 4;
    const int l16  = lane & 15;
    const int m0 = blockIdx.y * 64 + wave * 16;
    const int n0 = blockIdx.x * 64;

    const float* __restrict__ arow = A + (size_t)(m0 + l16) * THREEH;

    v8f acc[4] = {};
    for (int kc = 0; kc < THREEH; kc += 64) {
        __builtin_prefetch(arow + kc + 64, 0, 1);
        __builtin_prefetch(W + (size_t)(kc + 64 + lane) * Hdim + n0, 0, 1);
        __builtin_prefetch(W + (size_t)(kc + 96 + lane) * Hdim + n0, 0, 1);
#pragma unroll 4
        for (int k0 = kc; k0 < kc + 64; k0 += 4) {
            const int ka = k0 + half * 2;
            v2f a;
            a.x = arow[ka];
            a.y = arow[ka + 1];
            const float* bp = W + (size_t)ka * Hdim + n0 + l16;
#pragma unroll
            for (int t = 0; t < 4; ++t) {
                v2f b;
                b.x = bp[t * 16];
                b.y = bp[(size_t)Hdim + t * 16];
                acc[t] = wmma_f32(a, b, acc[t]);
            }
        }
    }
#pragma unroll
    for (int t = 0; t < 4; ++t) {
        const int col = n0 + t * 16 + l16;
#pragma unroll
        for (int r = 0; r < 8; ++r) {
            const int row = m0 + half * 8 + r;
            y[(size_t)row * Hdim + col] =
                acc[t][r] + bias[col] + hs[(size_t)row * Hdim + col];
        }
    }
}

// =====================================================================
// Kernel 4: RMSNorm rowwise: out = y * rsqrt(mean(y^2)+eps) * w
// =====================================================================
__global__ void __launch_bounds__(256)
k_rms(const float* __restrict__ y, const float* __restrict__ w,
      float* __restrict__ out) {
    __shared__ float red[8];
    const int row = blockIdx.x;
    const float* yr = y + (size_t)row * Hdim;
    float s = 0.0f;
    for (int i = threadIdx.x; i < Hdim; i += 256) { const float v = yr[i]; s += v * v; }
    for (int off = 16; off > 0; off >>= 1) s += __shfl_down(s, off, 32);
    const int wave = threadIdx.x >> 5;
    const int lane = threadIdx.x & 31;
    if (lane == 0) red[wave] = s;
    __syncthreads();
    if (wave == 0) {
        float t = (lane < 8) ? red[lane] : 0.0f;
        for (int off = 4; off > 0; off >>= 1) t += __shfl_down(t, off, 32);
        if (lane == 0) red[0] = t;
    }
    __syncthreads();
    const float rms = rsqrtf(red[0] * (1.0f / Hdim) + 1e-6f);
    for (int i = threadIdx.x; i < Hdim; i += 256)
        out[(size_t)row * Hdim + i] = yr[i] * rms * w[i];
}

// =====================================================================
extern "C" void kernel_launch(void* const* d_in, const int* in_sizes, int n_in,
                              void* d_out, int out_size, void* d_ws, size_t ws_size,
                              hipStream_t stream) {
    const float* hs  = (const float*)d_in[0];
    const int*   am  = (const int*)  d_in[1];
    const float* tsb = (const float*)d_in[2];
    const float* Wq  = (const float*)d_in[3];
    const float* Wo  = (const float*)d_in[4];
    const float* bo  = (const float*)d_in[5];
    const float* rw  = (const float*)d_in[6];

    float* fused = (float*)d_ws;                               // 4096 x 6144
    float* combg = fused + (size_t)MTOT * SIXH;                // 4096 x 3072
    float* ybuf  = combg + (size_t)MTOT * THREEH;              // 4096 x 1024
    float* out   = (float*)d_out;

    dim3 g1(SIXH / 64, MTOT / 64);
    k_qkvu<<<g1, 128, 0, stream>>>(hs, Wq, fused);

    dim3 g2(SEQ / 16, NH, Bsz);
    k_attn<<<g2, 32, 0, stream>>>(fused, am, tsb, combg);

    dim3 g3(Hdim / 64, MTOT / 64);
    k_outproj<<<g3, 128, 0, stream>>>(combg, Wo, bo, hs, ybuf);

    k_rms<<<MTOT, 256, 0, stream>>>(ybuf, rw, out);
}